// LinearCapsAR_43276090474892
// MI455X (gfx1250) — compile-verified
//
#include <hip/hip_runtime.h>
#include <math.h>

typedef __attribute__((ext_vector_type(2))) float v2f;
typedef __attribute__((ext_vector_type(4))) float v4f;
typedef __attribute__((ext_vector_type(8))) float v8f;
typedef __attribute__((ext_vector_type(4))) int   v4i;

static constexpr int Bsz  = 64;   // batch
static constexpr int Cin  = 32;   // in-channels  (K of the GEMM)
static constexpr int Din  = 256;  // in-capsules  (softmax axis)
static constexpr int Cout = 32;   // out-channels (N of the GEMM)
static constexpr int Dout = 256;  // out-capsules (one workgroup each)
static constexpr int XPITCH = 36; // padded LDS row (bank-conflict free, 16B aligned)
static constexpr int WPITCH = 36;

#if __has_builtin(__builtin_amdgcn_global_load_async_to_lds_b128)
#define USE_ASYNC_LDS 1
#else
#define USE_ASYNC_LDS 0
#endif

__device__ __forceinline__ void wait_async0() {
#if __has_builtin(__builtin_amdgcn_s_wait_asynccnt)
  __builtin_amdgcn_s_wait_asynccnt(0);
#else
  asm volatile("s_wait_asynccnt 0" ::: "memory");
#endif
}

// xs[D][b][C] = x[b][C][D]  -- makes per-D slabs contiguous for coalesced LDS fill
__global__ __launch_bounds__(256) void caps_transpose_kernel(
    const float* __restrict__ x, float* __restrict__ xs) {
  int o = blockIdx.x * blockDim.x + threadIdx.x;  // (D*64 + b)*32 + C
  int C = o & 31;
  int b = (o >> 5) & 63;
  int D = o >> 11;
  xs[o] = x[(b * Cin + C) * Din + D];
}

// 16-lane butterfly sum, VALU-only via DPP16 (no LDS traffic).
// After xor1+xor2 the value is uniform per 4-group, so row_half_mirror (lane^7)
// acts as xor4; after that it is uniform per 8-group, so row_mirror (lane^15)
// acts as xor8. Every lane of the 16-lane group ends with the full sum.
__device__ __forceinline__ float bfly16(float v) {
#if __has_builtin(__builtin_amdgcn_update_dpp)
  union { float f; int i; } u, r;
  u.f = v; r.i = __builtin_amdgcn_update_dpp(0, u.i, 0xB1,  0xf, 0xf, true); v += r.f; // quad_perm [1,0,3,2]
  u.f = v; r.i = __builtin_amdgcn_update_dpp(0, u.i, 0x4E,  0xf, 0xf, true); v += r.f; // quad_perm [2,3,0,1]
  u.f = v; r.i = __builtin_amdgcn_update_dpp(0, u.i, 0x141, 0xf, 0xf, true); v += r.f; // row_half_mirror
  u.f = v; r.i = __builtin_amdgcn_update_dpp(0, u.i, 0x140, 0xf, 0xf, true); v += r.f; // row_mirror
  return v;
#else
  v += __shfl_xor(v, 1, 16);
  v += __shfl_xor(v, 2, 16);
  v += __shfl_xor(v, 4, 16);
  v += __shfl_xor(v, 8, 16);
  return v;
#endif
}

__global__ __launch_bounds__(256) void caps_route_kernel(
    const float* __restrict__ xs,      // [Din][Bsz][Cin]
    const float* __restrict__ weight,  // [Dout][Din][Cout][Cin]
    const float* __restrict__ prior,   // [Dout][Din]
    float* __restrict__ out)           // [Bsz][Cout][Dout]
{
  // double-buffered slabs: [buf][j = D parity]
  __shared__ float xs_lds[2][2][Bsz * XPITCH];
  __shared__ float w_lds[2][2][Cout * WPITCH];
  __shared__ float prior_lds[Din];
  __shared__ float sm_acc[4][16][32];         // flash combine scratch per m-tile
  __shared__ float sm_max[4][16];
  __shared__ float sm_l[4][16];

  const int d      = blockIdx.x;
  const int t      = threadIdx.x;
  const int wave   = t >> 5;        // 0..7
  const int lane   = t & 31;
  const int mtile  = wave & 3;      // which 16 batch rows
  const int parity = wave >> 2;     // which D parity this wave owns
  const int lh     = lane >> 4;     // lane half (selects K pair / M+8)
  const int lm     = lane & 15;     // M row (A) / N col (B,C)

  if (t < 64) {  // preload prior[d, :]
    *(v4f*)&prior_lds[t * 4] = *(const v4f*)(prior + (size_t)d * Din + t * 4);
  }

  const float* wslab = weight + (size_t)d * Din * Cout * Cin;

  // per-thread staging geometry (same for both paths)
  const int xm0 = t >> 3,          xc0 = (t & 7) << 2;          // x float4 #t
  const int xm1 = (t + 256) >> 3,  xc1 = ((t + 256) & 7) << 2;  // x float4 #t+256
  const int wn  = t >> 3,          wc  = (t & 7) << 2;          // w float4 #t

  v8f acc0 = {}; v8f acc1 = {};          // running att-weighted sums (n = lm, lm+16)
  float mmax[8], lsum[8];                // online-softmax state, row = i + 8*lh
#pragma unroll
  for (int i = 0; i < 8; ++i) { mmax[i] = -INFINITY; lsum[i] = 0.0f; }

  const float invNC = 0.17677669529663687f;  // 1/sqrt(32)

#if USE_ASYNC_LDS
  // -------- async DMA path: global -> LDS directly, tracked by ASYNCcnt ----
  typedef __attribute__((address_space(1))) v4i gv4i_t;  // global int4
  typedef __attribute__((address_space(3))) v4i lv4i_t;  // LDS int4
  auto issue_async = [&](int buf, int Dbase) {
#pragma unroll
    for (int j = 0; j < 2; ++j) {
      const float* xsrc = xs + (size_t)(Dbase + j) * (Bsz * Cin);
      __builtin_amdgcn_global_load_async_to_lds_b128(
          (gv4i_t*)(xsrc + t * 4),
          (lv4i_t*)&xs_lds[buf][j][xm0 * XPITCH + xc0], 0, 0);
      __builtin_amdgcn_global_load_async_to_lds_b128(
          (gv4i_t*)(xsrc + (t + 256) * 4),
          (lv4i_t*)&xs_lds[buf][j][xm1 * XPITCH + xc1], 0, 0);
      const float* wsrc = wslab + (size_t)(Dbase + j) * (Cout * Cin);
      __builtin_amdgcn_global_load_async_to_lds_b128(
          (gv4i_t*)(wsrc + t * 4),
          (lv4i_t*)&w_lds[buf][j][wn * WPITCH + wc], 0, 0);
    }
  };
  issue_async(0, 0);
#else
  // -------- fallback: register-staged, software-pipelined ------------------
  v4f xr[2][2], wr[2];
  auto load_slabs = [&](int Dbase) {
#pragma unroll
    for (int j = 0; j < 2; ++j) {
      const float* xsrc = xs + (size_t)(Dbase + j) * (Bsz * Cin);
      xr[j][0] = *(const v4f*)&xsrc[t * 4];
      xr[j][1] = *(const v4f*)&xsrc[(t + 256) * 4];
      const float* wsrc = wslab + (size_t)(Dbase + j) * (Cout * Cin);
      wr[j] = *(const v4f*)&wsrc[t * 4];
    }
  };
  load_slabs(0);
#endif

  for (int Dbase = 0; Dbase < Din; Dbase += 2) {
    const int buf = (Dbase >> 1) & 1;

#if USE_ASYNC_LDS
    wait_async0();                 // my prefetch for THIS buffer has landed
    __syncthreads();               // everyone's landed; prev compute (buf^1) done
    if (Dbase + 2 < Din) issue_async(buf ^ 1, Dbase + 2);  // overlaps compute
#else
    __syncthreads();               // previous compute done; LDS reusable
#pragma unroll
    for (int j = 0; j < 2; ++j) {
      *(v4f*)&xs_lds[buf][j][xm0 * XPITCH + xc0] = xr[j][0];
      *(v4f*)&xs_lds[buf][j][xm1 * XPITCH + xc1] = xr[j][1];
      *(v4f*)&w_lds[buf][j][wn * WPITCH + wc]    = wr[j];
    }
    __syncthreads();               // LDS filled
    if (Dbase + 2 < Din) load_slabs(Dbase + 2);  // prefetch next, overlaps compute
#endif

    // --- this wave's D: u[16 x 32] = X[16 x 32] * W^T via 16 x (16x16x4) f32 WMMA
    const float* xrow = &xs_lds[buf][parity][(mtile * 16 + lm) * XPITCH];
    const float* wn0  = &w_lds[buf][parity][lm * WPITCH];
    const float* wn1  = &w_lds[buf][parity][(16 + lm) * WPITCH];
    v8f u0 = {}; v8f u1 = {};
#pragma unroll
    for (int kk = 0; kk < 8; ++kk) {
      const int ko = kk * 4 + lh * 2;     // A/B hold K pairs: lanes0-15 K0K1, 16-31 K2K3
      v2f a  = *(const v2f*)&xrow[ko];
      v2f b0 = *(const v2f*)&wn0[ko];
      v2f b1 = *(const v2f*)&wn1[ko];
      u0 = __builtin_amdgcn_wmma_f32_16x16x4_f32(false, a, false, b0, (short)0, u0, false, false);
      u1 = __builtin_amdgcn_wmma_f32_16x16x4_f32(false, a, false, b1, (short)0, u1, false, false);
    }

    const float pr = prior_lds[Dbase + parity];

    // --- per-row score (||u||^2 over 32 c) + online softmax update
#pragma unroll
    for (int i = 0; i < 8; ++i) {
      float q = bfly16(u0[i] * u0[i] + u1[i] * u1[i]);  // every lane: score[row]
      float logit = q * invNC + pr;
      float newm  = fmaxf(mmax[i], logit);
      float scale = __expf(mmax[i] - newm);
      float p     = __expf(logit - newm);
      lsum[i] = lsum[i] * scale + p;
      acc0[i] = acc0[i] * scale + p * u0[i];
      acc1[i] = acc1[i] * scale + p * u1[i];
      mmax[i] = newm;
    }
  }

  // --- flash combine across the two parity waves of each m-tile
  if (parity == 0) {
#pragma unroll
    for (int i = 0; i < 8; ++i) {
      int row = i + 8 * lh;
      sm_acc[mtile][row][lm]      = acc0[i];
      sm_acc[mtile][row][lm + 16] = acc1[i];
      if (lm == 0) { sm_max[mtile][row] = mmax[i]; sm_l[mtile][row] = lsum[i]; }
    }
  }
  __syncthreads();
  if (parity == 1) {
#pragma unroll
    for (int i = 0; i < 8; ++i) {
      int row = i + 8 * lh;
      float m0 = sm_max[mtile][row];
      float l0 = sm_l[mtile][row];
      float M  = fmaxf(m0, mmax[i]);
      float s0 = __expf(m0 - M);
      float s1 = __expf(mmax[i] - M);
      float L  = l0 * s0 + lsum[i] * s1;
      float ri = 1.0f / L;
      float r0 = (sm_acc[mtile][row][lm]      * s0 + acc0[i] * s1) * ri;
      float r1 = (sm_acc[mtile][row][lm + 16] * s0 + acc1[i] * s1) * ri;
      int b = mtile * 16 + row;
      out[((size_t)b * Cout + lm)      * Dout + d] = r0;
      out[((size_t)b * Cout + lm + 16) * Dout + d] = r1;
    }
  }
}

extern "C" void kernel_launch(void* const* d_in, const int* in_sizes, int n_in,
                              void* d_out, int out_size, void* d_ws, size_t ws_size,
                              hipStream_t stream) {
  const float* x     = (const float*)d_in[0];  // (64, 32, 256)
  const float* w     = (const float*)d_in[1];  // (256, 256, 32, 32)
  const float* prior = (const float*)d_in[2];  // (1, 256, 256)
  float* outp = (float*)d_out;                 // (64, 32, 256)
  float* xsws = (float*)d_ws;                  // needs 2 MB

  caps_transpose_kernel<<<(Bsz * Cin * Din) / 256, 256, 0, stream>>>(x, xsws);
  caps_route_kernel<<<Dout, 256, 0, stream>>>(xsws, w, prior, outp);
}